// AlignerOT_91259465106076
// MI455X (gfx1250) — compile-verified
//
#include <hip/hip_runtime.h>

#define D      128
#define BS     2048      // B*S = 8*256
#define PPB    8         // problems per block in the sinkhorn kernel
#define NITER  20
#define SCALE  300.0f
#define EPS    0.1f

typedef __attribute__((ext_vector_type(2))) float v2f;
typedef __attribute__((ext_vector_type(8))) float v8f;

// ---------------------------------------------------------------------------
// Zero the [128,128] plan accumulator (workspace is poisoned, not zeroed).
// ---------------------------------------------------------------------------
__global__ void zero_acc_kernel(float* __restrict__ acc) {
  int i = blockIdx.x * blockDim.x + threadIdx.x;
  if (i < D * D) acc[i] = 0.0f;
}

// ---------------------------------------------------------------------------
// One block (128 threads, 4 wave32) processes PPB=8 (b,s) problems serially.
// phi_m = log(1/128) - LSE_n(A[m,n] + psi_n),  A = -3000*(x_m - y_n)^2
// psi_n = log(1/128) - LSE_m(A[m,n] + phi_m)
// plan  = exp(A + phi + psi), accumulated into a 64KB LDS accumulator
// (transposed accs[n*D+m] so lanes hit consecutive banks), flushed once per
// block with agent-scope f32 atomics -> 8x fewer global atomics.
// A is recomputed from x,y (3 VALU ops) instead of materializing 64KB.
// LDS/block = 67.6KB -> 4 blocks (16 waves) per 320KB WGP.
// ---------------------------------------------------------------------------
__global__ __launch_bounds__(128) void sinkhorn_kernel(const float* __restrict__ X,
                                                       const float* __restrict__ Y,
                                                       float* __restrict__ acc) {
  const int t = threadIdx.x;   // 0..127 : row index (phi phase) / col index (psi phase)

  __shared__ float xsh[D], ysh[D], phi[D], psi[D];
  __shared__ float accs[D * D];            // accs[n*D + m], 64KB

  // zero LDS accumulator (covered by the first barrier below)
  for (int i = t; i < D * D; i += D) accs[i] = 0.0f;

  const float NSE   = -(SCALE / EPS);          // -3000
  const float LOGMU = -4.852030263919617f;     // -ln(128), also log nu

  const int p0 = blockIdx.x * PPB;
  for (int pp = 0; pp < PPB; ++pp) {
    const int p = p0 + pp;
    __syncthreads();   // prev accumulate (reads ysh/psi) done; accs zeroed (pp==0)

    const float xm = X[(size_t)p * D + t];
    const float yn = Y[(size_t)p * D + t];
    xsh[t] = xm;
    ysh[t] = yn;
    psi[t] = 0.0f;
    if (pp + 1 < PPB) {                       // hide next problem's latency
      __builtin_prefetch(&X[(size_t)(p + 1) * D + t], 0, 0);   // global_prefetch_b8
      __builtin_prefetch(&Y[(size_t)(p + 1) * D + t], 0, 0);
    }
    __syncthreads();

    for (int it = 0; it < NITER; ++it) {
      // ---- phi update: thread t owns row m = t; LSE over n of A + psi[n] ----
      float mx = -3.0e38f;
#pragma unroll 4
      for (int n = 0; n < D; ++n) {
        float d = xm - ysh[n];
        float a = fmaf(NSE * d, d, psi[n]);
        mx = fmaxf(mx, a);
      }
      float s = 0.0f;
#pragma unroll 4
      for (int n = 0; n < D; ++n) {
        float d = xm - ysh[n];
        float a = fmaf(NSE * d, d, psi[n]);
        s += __expf(a - mx);
      }
      phi[t] = LOGMU - (mx + __logf(s));   // distinct array: no hazard vs psi reads
      __syncthreads();

      // ---- psi update: thread t owns col n = t; LSE over m of A + phi[m] ----
      mx = -3.0e38f;
#pragma unroll 4
      for (int m = 0; m < D; ++m) {
        float d = xsh[m] - yn;
        float a = fmaf(NSE * d, d, phi[m]);
        mx = fmaxf(mx, a);
      }
      s = 0.0f;
#pragma unroll 4
      for (int m = 0; m < D; ++m) {
        float d = xsh[m] - yn;
        float a = fmaf(NSE * d, d, phi[m]);
        s += __expf(a - mx);
      }
      psi[t] = LOGMU - (mx + __logf(s));
      __syncthreads();
    }

    // ---- accumulate transport plan row m=t into LDS (thread-private cells) ----
    const float ph = phi[t];
#pragma unroll 4
    for (int n = 0; n < D; ++n) {
      float d = xm - ysh[n];
      float a = fmaf(NSE * d, d, ph + psi[n]);
      accs[n * D + t] += __expf(a);        // lane-consecutive -> conflict-free
    }
  }

  __syncthreads();
  // ---- flush LDS accumulator to global with one atomic per element ----
  for (int i = t; i < D * D; i += D) {
    const int n = i >> 7;         // accs layout is [n][m]
    const int m = i & (D - 1);
    __hip_atomic_fetch_add(&acc[m * D + n], accs[i],
                           __ATOMIC_RELAXED, __HIP_MEMORY_SCOPE_AGENT);
  }
}

// ---------------------------------------------------------------------------
// ot[m,n] = acc[m,n] * (SCALE * n / (B*S)) + delta_ot[m,n] = acc * 18.75 + delta
// ---------------------------------------------------------------------------
__global__ void finalize_kernel(const float* __restrict__ acc,
                                const float* __restrict__ delta,
                                float* __restrict__ ot) {
  int i = blockIdx.x * blockDim.x + threadIdx.x;
  if (i < D * D) {
    const float f = (SCALE * (float)D) / (float)BS;   // 18.75
    ot[i] = fmaf(acc[i], f, delta[i]);
  }
}

// ---------------------------------------------------------------------------
// out[2048,128] = X[2048,128] @ ot[128,128] in exact f32 via
// V_WMMA_F32_16X16X4_F32. One wave per 16x16 output tile; 8 waves/block cover
// N=128; 128 blocks cover M=2048. 32 WMMA ops per tile (K=128/4).
// Fragment layout per ISA 7.12.2:
//   A (16x4):  lane holds A[M][K0], A[M][K0+1], M=lane&15, K0=2*(lane>>4)
//   B (4x16):  lane holds B[K0][N], B[K0+1][N], N=lane&15
//   C/D 16x16: VGPR r holds row (r + 8*(lane>>4)), col = lane&15
// ---------------------------------------------------------------------------
__global__ __launch_bounds__(256) void gemm_wmma_kernel(const float* __restrict__ Xf,
                                                        const float* __restrict__ ot,
                                                        float* __restrict__ out) {
  const int lane  = threadIdx.x & 31;
  const int wid   = threadIdx.x >> 5;
  const int l16   = lane & 15;
  const int k0l   = (lane >> 4) * 2;
  const int mBase = blockIdx.x * 16;
  const int nBase = wid * 16;

  __builtin_prefetch(&ot[(size_t)k0l * D + nBase + l16], 0, 0);  // warm L2 path

  v8f c = {};  // zero C accumulator

  const float* arow = Xf + (size_t)(mBase + l16) * D;

#pragma unroll 4
  for (int k = 0; k < D; k += 4) {
    v2f a, b;
    a.x = arow[k + k0l];                                  // contiguous -> b64 load
    a.y = arow[k + k0l + 1];
    b.x = ot[(size_t)(k + k0l)     * D + nBase + l16];
    b.y = ot[(size_t)(k + k0l + 1) * D + nBase + l16];
    // (neg_a, A, neg_b, B, c_mod, C, reuse_a, reuse_b)
    c = __builtin_amdgcn_wmma_f32_16x16x4_f32(false, a, false, b,
                                              (short)0, c, false, false);
  }

  const int rowBase = mBase + (lane >> 4) * 8;
#pragma unroll
  for (int r = 0; r < 8; ++r) {
    out[(size_t)(rowBase + r) * D + nBase + l16] = c[r];
  }
}

// ---------------------------------------------------------------------------
extern "C" void kernel_launch(void* const* d_in, const int* in_sizes, int n_in,
                              void* d_out, int out_size, void* d_ws, size_t ws_size,
                              hipStream_t stream) {
  const float* X     = (const float*)d_in[0];   // [8,256,128]
  const float* Y     = (const float*)d_in[1];   // [8,256,128]
  const float* delta = (const float*)d_in[2];   // [128,128]
  float*       out   = (float*)d_out;           // [8,256,128]

  float* acc = (float*)d_ws;                    // [128,128] plan-sum accumulator
  float* ot  = acc + D * D;                     // [128,128] final ot_matrix

  zero_acc_kernel<<<(D * D + 255) / 256, 256, 0, stream>>>(acc);
  sinkhorn_kernel<<<BS / PPB, D, 0, stream>>>(X, Y, acc);
  finalize_kernel<<<(D * D + 255) / 256, 256, 0, stream>>>(acc, delta, ot);
  gemm_wmma_kernel<<<BS / 16, 256, 0, stream>>>(X, ot, out);
}